// NSF_AR_52261162058415
// MI455X (gfx1250) — compile-verified
//
#include <hip/hip_runtime.h>
#include <math.h>

typedef float v2f __attribute__((ext_vector_type(2)));
typedef float v8f __attribute__((ext_vector_type(8)));

#define NS 32768            // samples
#define DD 64               // dims
#define LL 63               // layers (d-1)
#define PP 14               // 3K-1
#define NCAT 512            // 63*8 padded to 512
#define BB 5.0f
#define MINW 0.001f
#define MINH 0.001f
#define MIND 0.001f

// ---------------- pack kernel: build Bt[col][k], col = l*8+h (transposed, zero-padded, masked)
__global__ void nsf_pack_w1(const float* __restrict__ W1, float* __restrict__ Bt) {
    int idx = blockIdx.x * blockDim.x + threadIdx.x;   // 0 .. 512*64-1
    if (idx >= NCAT * 64) return;
    int col = idx >> 6;          // 0..511
    int k   = idx & 63;          // 0..63
    int l   = col >> 3;          // layer
    int h   = col & 7;           // hidden unit
    float v = 0.0f;
    if (l < LL && k <= l) v = W1[(l * LL + k) * 8 + h];   // triangular mask implicit
    Bt[col * 64 + k] = v;
}

// ---------------- rational-quadratic spline (elementwise, fp32, matches reference math)
__device__ __forceinline__ float softplusf(float v) {
    return v > 20.0f ? v : log1pf(expf(v));
}

__device__ void rqs_eval(const float* __restrict__ p, float xf, float& z, float& ld) {
    float cumw[6], cumh[6], derivs[6];
    // widths -> cumw
    {
        float mx = p[0];
        #pragma unroll
        for (int i = 1; i < 5; i++) mx = fmaxf(mx, p[i]);
        float e[5], s = 0.0f;
        #pragma unroll
        for (int i = 0; i < 5; i++) { e[i] = expf(p[i] - mx); s += e[i]; }
        float inv = 1.0f / s, cs = 0.0f;
        cumw[0] = -BB;
        #pragma unroll
        for (int i = 0; i < 4; i++) {
            cs += MINW + (1.0f - MINW * 5.0f) * e[i] * inv;
            cumw[i + 1] = 2.0f * BB * cs - BB;
        }
        cumw[5] = BB;
    }
    // heights -> cumh
    {
        float mx = p[5];
        #pragma unroll
        for (int i = 6; i < 10; i++) mx = fmaxf(mx, p[i]);
        float e[5], s = 0.0f;
        #pragma unroll
        for (int i = 0; i < 5; i++) { e[i] = expf(p[5 + i] - mx); s += e[i]; }
        float inv = 1.0f / s, cs = 0.0f;
        cumh[0] = -BB;
        #pragma unroll
        for (int i = 0; i < 4; i++) {
            cs += MINH + (1.0f - MINH * 5.0f) * e[i] * inv;
            cumh[i + 1] = 2.0f * BB * cs - BB;
        }
        cumh[5] = BB;
    }
    // derivatives: padded ends give softplus(const) = 1-MIND -> deriv = 1.0
    derivs[0] = 1.0f; derivs[5] = 1.0f;
    #pragma unroll
    for (int i = 1; i < 5; i++) derivs[i] = MIND + softplusf(p[9 + i]);

    float xc = fminf(fmaxf(xf, -BB), BB);
    int idx = 0;
    #pragma unroll
    for (int j = 1; j < 6; j++) idx += (xc >= cumw[j]) ? 1 : 0;
    if (idx > 4) idx = 4;

    float icw = cumw[idx], ibw = cumw[idx + 1] - cumw[idx];
    float ich = cumh[idx], ih  = cumh[idx + 1] - cumh[idx];
    float idl = ih / ibw;
    float idv = derivs[idx], idvp1 = derivs[idx + 1];
    float theta = (xc - icw) / ibw;
    float tmt = theta * (1.0f - theta);
    float num = ih * (idl * theta * theta + idv * tmt);
    float den = idl + (idv + idvp1 - 2.0f * idl) * tmt;
    float outv = ich + num / den;
    float omt = 1.0f - theta;
    float dnum = idl * idl * (idvp1 * theta * theta + 2.0f * idl * tmt + idv * omt * omt);
    float ldv = logf(dnum) - 2.0f * logf(den);
    bool inside = (xf >= -BB) && (xf <= BB);
    z  = inside ? outv : xf;
    ld = inside ? ldv : 0.0f;
}

// ---------------- fused main kernel: WMMA layer1 + VALU layers 2/3 + spline
__launch_bounds__(128)
__global__ void nsf_fused(const float* __restrict__ x, const float* __restrict__ initp,
                          const float* __restrict__ b1, const float* __restrict__ W2,
                          const float* __restrict__ b2, const float* __restrict__ W3,
                          const float* __restrict__ b3, const float* __restrict__ Bt,
                          float* __restrict__ out) {
    __shared__ float tile[4][16][17];   // per-wave 16x16 h tile (padded)
    __shared__ float ldacc[DD];         // per-column logdet sum over this block's 64 samples

    const int tid  = threadIdx.x;
    const int wave = tid >> 5;
    const int lane = tid & 31;
    const int m    = lane & 15;        // sample within wave tile / N within WMMA tile
    const int hh   = lane >> 4;        // K-half selector / layer parity
    const int s0   = blockIdx.x * 64 + wave * 16;
    const int s    = s0 + m;           // this lane's global sample

    if (tid < DD) ldacc[tid] = 0.0f;
    __syncthreads();

    // column 0: spline with init_param (lanes 0..15 of each wave, one sample each)
    if (lane < 16) {
        float p[PP];
        #pragma unroll
        for (int q = 0; q < PP; q++) p[q] = initp[q];
        float z, ld;
        rqs_eval(p, x[s * DD + 0], z, ld);
        out[s] = z;                               // c = 0
        atomicAdd(&ldacc[0], ld);
    }

    // preload A fragments: lane m holds row s; half hh selects K pair per 4-step
    v2f a[16];
    const float* xrow = x + (size_t)s * DD;
    #pragma unroll
    for (int kk = 0; kk < 16; kk++)
        a[kk] = *(const v2f*)(xrow + 4 * kk + 2 * hh);

    for (int t = 0; t < 32; ++t) {
        // ---- layer 1: 16x16x64 via 16 accumulating v_wmma_f32_16x16x4_f32
        const float* bcol = Bt + (size_t)(16 * t + m) * 64 + 2 * hh;
        v8f c = {0.f, 0.f, 0.f, 0.f, 0.f, 0.f, 0.f, 0.f};
        #pragma unroll
        for (int kk = 0; kk < 16; kk++) {
            v2f b = *(const v2f*)(bcol + 4 * kk);
            c = __builtin_amdgcn_wmma_f32_16x16x4_f32(
                    false, a[kk], false, b, (short)0, c, false, false);
        }

        // ---- bias + tanh; D layout: VGPR r, lane -> (M = r + 8*hh, N = m)
        const int col = 16 * t + m;                 // N within W1cat
        const float bias = (col < LL * 8) ? b1[col] : 0.0f;
        #pragma unroll
        for (int r = 0; r < 8; r++) {
            float hv = tanhf(c[r] + bias);
            tile[wave][r + 8 * hh][m] = hv;
        }
        __syncthreads();

        // ---- layers 2/3 + spline; lane = (sample m, layer parity hh)
        const int l_idx = 2 * t + hh;               // 0..63; 63 is padding
        if (l_idx < LL) {
            float hv[8];
            #pragma unroll
            for (int j = 0; j < 8; j++) hv[j] = tile[wave][m][8 * hh + j];

            const float* w2  = W2 + (size_t)l_idx * 64;
            const float* bb2 = b2 + (size_t)l_idx * 8;
            float h2[8];
            #pragma unroll
            for (int g = 0; g < 8; g++) {
                float acc = bb2[g];
                #pragma unroll
                for (int j = 0; j < 8; j++) acc += hv[j] * w2[j * 8 + g];
                h2[g] = tanhf(acc);
            }

            const float* w3  = W3 + (size_t)l_idx * 8 * PP;
            const float* bb3 = b3 + (size_t)l_idx * PP;
            float pv[PP];
            #pragma unroll
            for (int q = 0; q < PP; q++) {
                float acc = bb3[q];
                #pragma unroll
                for (int g = 0; g < 8; g++) acc += h2[g] * w3[g * PP + q];
                pv[q] = acc;
            }

            const int cdim = l_idx + 1;             // spline column
            float z, ld;
            rqs_eval(pv, x[(size_t)s * DD + cdim], z, ld);
            out[(size_t)cdim * NS + s] = z;
            atomicAdd(&ldacc[cdim], ld);
        }
        __syncthreads();
    }

    // logdet outputs: reference's reshape(n,d).sum(1) groups 64 consecutive samples of one column.
    // out[n*d + c*512 + blockIdx.x] = sum over this block's 64 samples of ld(column c).
    if (tid < DD)
        out[(size_t)NS * DD + (size_t)tid * (NS / 64) + blockIdx.x] = ldacc[tid];
}

extern "C" void kernel_launch(void* const* d_in, const int* in_sizes, int n_in,
                              void* d_out, int out_size, void* d_ws, size_t ws_size,
                              hipStream_t stream) {
    const float* x     = (const float*)d_in[0];
    const float* initp = (const float*)d_in[1];
    const float* W1    = (const float*)d_in[2];
    const float* b1    = (const float*)d_in[3];
    const float* W2    = (const float*)d_in[4];
    const float* b2    = (const float*)d_in[5];
    const float* W3    = (const float*)d_in[6];
    const float* b3    = (const float*)d_in[7];
    float* out = (float*)d_out;
    float* Bt  = (float*)d_ws;                     // 512*64*4 = 128 KB packed W1cat^T

    nsf_pack_w1<<<(NCAT * 64 + 255) / 256, 256, 0, stream>>>(W1, Bt);
    nsf_fused<<<NS / 64, 128, 0, stream>>>(x, initp, b1, W2, b2, W3, b3, Bt, out);
}